// BiLinearInteractionLayer_34471407518397
// MI455X (gfx1250) — compile-verified
//
#include <hip/hip_runtime.h>

// ---------------------------------------------------------------------------
// BiLinearInteractionLayer on gfx1250 (CDNA5):
//   out[b,p,e] = sum_d x[b, I(p), d] * W[p, e, d]  *  x[b, J(p), e]
// 496 independent 4096x64 @ 64x64 f32 GEMMs + elementwise epilogue.
// Bandwidth-bound on the 520 MB f32 output (~24 us floor at 23.3 TB/s), so:
//  - V_WMMA_F32_16X16X4_F32 for the GEMM (full f32, compute far under floor)
//  - xj operands preloaded up-front so their latency hides under the WMMAs
//  - non-temporal output stores: write-once stream must not evict the
//    reused x (33 MB) / W (8 MB) working set from the 192 MB L2
// ---------------------------------------------------------------------------

#define NF        32
#define DIMK      64
#define NPAIR     496
#define BATCH     4096
#define BATCH_TILE 128            // 8 waves * 16 rows each
#define LDS_PITCH 68              // 64 + 4 pad: bank-conflict-free ds_read_b64,
                                  // 16B-aligned rows for v4f stores

typedef __attribute__((ext_vector_type(2))) float v2f;
typedef __attribute__((ext_vector_type(4))) float v4f;
typedef __attribute__((ext_vector_type(8))) float v8f;

__global__ __launch_bounds__(256, 2)
void bilinear_wmma_f32_kernel(const float* __restrict__ x,
                              const float* __restrict__ W,
                              float* __restrict__ out)
{
    __shared__ float sW[DIMK * LDS_PITCH];   // 17,408 B of the 320 KB WGP LDS

    const int p = blockIdx.y;

    // Decode lexicographic pair index p -> (i, j), i < j. Scalar/uniform.
    int i = 0, rem = p;
    while (rem >= NF - 1 - i) { rem -= NF - 1 - i; ++i; }
    const int j = i + 1 + rem;

    const int tid = threadIdx.x;

    // ---- Stage W[p] (64x64 f32, row-major over [e][d]) into padded LDS ----
    {
        const float* Wp = W + (size_t)p * (DIMK * DIMK);
        #pragma unroll
        for (int it = 0; it < 4; ++it) {
            const int idx = tid + it * 256;     // 1024 float4 chunks total
            const int e  = idx >> 4;            // 16 float4 per 64-float row
            const int d4 = idx & 15;
            const v4f v = *(const v4f*)(Wp + e * DIMK + d4 * 4);
            *(v4f*)(sW + e * LDS_PITCH + d4 * 4) = v;   // 272B pitch: 16B aligned
        }
    }

    // ---- Per-wave 16x64 output tile ----
    const int wave = tid >> 5;          // 0..7
    const int lane = tid & 31;
    const int half = lane >> 4;         // 0: lanes 0-15, 1: lanes 16-31
    const int ln   = lane & 15;

    const int b0 = blockIdx.x * BATCH_TILE + wave * 16;

    const size_t out_row_stride = (size_t)NPAIR * DIMK;            // per batch row
    const size_t x_row_stride   = (size_t)NF * DIMK;

    // A fragments: 16x4 f32 layout. Lane (half, ln) supplies row m = ln,
    // K = {4k + 2*half, 4k + 2*half + 1} for chunk k.
    const float* xrow_i = x + ((size_t)(b0 + ln) * NF + i) * DIMK + 2 * half;
    v2f afrag[16];
    #pragma unroll
    for (int k = 0; k < 16; ++k)
        afrag[k] = *(const v2f*)(xrow_i + 4 * k);

    // xj operands for the epilogue, all issued now so the 64 WMMAs below hide
    // their latency. C/D layout: VGPR r -> M = r + 8*half, N = ln, e = 16t+ln.
    // Each (t, r) group is a 64B-coalesced fetch per half-wave.
    const float* xj_base = x + ((size_t)b0 + 8 * half) * x_row_stride
                             + (size_t)j * DIMK + ln;
    float xjv[4][8];
    #pragma unroll
    for (int t = 0; t < 4; ++t)
        #pragma unroll
        for (int r = 0; r < 8; ++r)
            xjv[t][r] = xj_base[(size_t)r * x_row_stride + t * 16];

    __syncthreads();

    #pragma unroll
    for (int t = 0; t < 4; ++t) {                  // 4 N-tiles of 16 output dims
        const int e = t * 16 + ln;                 // this lane's output column

        // B fragments from LDS: B[K][N] = W[N][K]; lane (half, ln) supplies
        // column N = e, K = {4k + 2*half, 4k + 2*half + 1}.
        const float* wbase = sW + e * LDS_PITCH + 2 * half;

        v8f acc = {};
        #pragma unroll
        for (int k = 0; k < 16; ++k) {
            const v2f bfrag = *(const v2f*)(wbase + 4 * k);
            acc = __builtin_amdgcn_wmma_f32_16x16x4_f32(
                      /*neg_a=*/false, afrag[k],
                      /*neg_b=*/false, bfrag,
                      /*c_mod=*/(short)0, acc,
                      /*reuse_a=*/false, /*reuse_b=*/false);
        }

        // Epilogue: pure mul + streaming (non-temporal) store, no load waits.
        float* out_base = out + ((size_t)b0 + 8 * half) * out_row_stride
                              + (size_t)p * DIMK + e;
        #pragma unroll
        for (int r = 0; r < 8; ++r)
            __builtin_nontemporal_store(acc[r] * xjv[t][r],
                                        out_base + (size_t)r * out_row_stride);
    }
}

extern "C" void kernel_launch(void* const* d_in, const int* in_sizes, int n_in,
                              void* d_out, int out_size, void* d_ws, size_t ws_size,
                              hipStream_t stream) {
    (void)in_sizes; (void)n_in; (void)d_ws; (void)ws_size; (void)out_size;
    const float* x = (const float*)d_in[0];   // (4096, 32, 64) f32
    const float* W = (const float*)d_in[1];   // (496, 64, 64) f32
    float* out = (float*)d_out;               // (4096, 496, 64) f32

    dim3 grid(BATCH / BATCH_TILE, NPAIR, 1);  // (32, 496)
    bilinear_wmma_f32_kernel<<<grid, 256, 0, stream>>>(x, W, out);
}